// FractionalPhysicsLoss_48661979464033
// MI455X (gfx1250) — compile-verified
//
#include <hip/hip_runtime.h>
#include <math.h>

// ---------------------------------------------------------------------------
// FractionalPhysicsLoss for MI455X (gfx1250, wave32, WMMA).
//
// physics term: per-row causal conv y[n] = sum_{k<=n} b[k] du[n-k]
//   == [R x K] x [K x K lower-triangular Toeplitz] matmul
//   -> tiled V_WMMA_F32_16X16X4_F32.
//   du slab in LDS (pitch 1028 dwords => conflict-free ds_load_b64 for the
//   16 A rows). Toeplitz coefficients stored REVERSED with guard zeros so
//   (b[d], b[d-1]) is an ascending consecutive LDS pair (no swap movs) and
//   diagonal-tile masking needs no predication. Each wave accumulates FOUR
//   output tiles at once so one A load feeds four WMMAs in the shared
//   region of the triangle.
// ---------------------------------------------------------------------------

typedef __attribute__((ext_vector_type(2))) float v2f;
typedef __attribute__((ext_vector_type(8))) float v8f;

#define ROWLEN 1024          // N
#define KLEN   1023          // N-1 (du length / Toeplitz size)
#define NROW   16            // rows per block (WMMA M)
#define PITCH  1028          // LDS row pitch in floats (bank-conflict-free)
#define BLEN   1040          // reversed coeffs + guard zeros
#define NELTS  8388608       // B*T*N

__global__ void fpl_init_kernel(double* acc) {
    acc[0] = 0.0;   // data-loss accumulator
    acc[1] = 0.0;   // physics-loss accumulator
}

// ---- data loss: mean((u_pred - u_true)^2), bandwidth bound -----------------
__global__ __launch_bounds__(256) void fpl_data_kernel(
        const float* __restrict__ a, const float* __restrict__ b,
        int n, double* __restrict__ acc) {
    const int idx    = blockIdx.x * blockDim.x + threadIdx.x;
    const int stride = gridDim.x * blockDim.x;
    const float4* a4 = (const float4*)a;
    const float4* b4 = (const float4*)b;
    float s = 0.0f;
    for (int i = idx; i < n / 4; i += stride) {
        float4 x = a4[i];
        float4 y = b4[i];
        float d0 = x.x - y.x, d1 = x.y - y.y, d2 = x.z - y.z, d3 = x.w - y.w;
        s += d0 * d0 + d1 * d1 + d2 * d2 + d3 * d3;
    }
    for (int off = 16; off > 0; off >>= 1) s += __shfl_down(s, off);
    __shared__ float wsum[8];
    const int lane = threadIdx.x & 31;
    const int w    = threadIdx.x >> 5;
    if (lane == 0) wsum[w] = s;
    __syncthreads();
    if (threadIdx.x < 8) {
        float t = wsum[threadIdx.x];
        for (int off = 4; off > 0; off >>= 1) t += __shfl_down(t, off);
        if (threadIdx.x == 0) atomicAdd(&acc[0], (double)t);
    }
}

// ---- physics loss: WMMA Toeplitz matmul + residual -------------------------
__global__ __launch_bounds__(128) void fpl_physics_kernel(
        const float* __restrict__ u, double* __restrict__ acc) {
    extern __shared__ float smem[];
    float* du_s = smem;                  // NROW * PITCH
    float* brv  = smem + NROW * PITCH;   // BLEN: brv[x] = b[1023-x] (else 0)

    const int tid = threadIdx.x;
    const int rb  = blockIdx.x * NROW;   // first global row of this slab

    // reversed coefficients: b[k] = (k+1)^(1-alpha) - k^(1-alpha), alpha=0.5
    // brv[x] = b[1023-x] for 1 <= x <= 1023, zero guards elsewhere
    for (int x = tid; x < BLEN; x += 128) {
        int k = 1023 - x;
        float kf = (float)k;
        brv[x] = (k >= 0 && k < KLEN) ? (sqrtf(kf + 1.0f) - sqrtf(kf)) : 0.0f;
    }
    // du rows into LDS (padding zeroed so tail tiles stay finite)
    for (int r = 0; r < NROW; ++r) {
        const float* up = u + (size_t)(rb + r) * ROWLEN;
        float* dr = du_s + r * PITCH;
        for (int m = tid; m < PITCH; m += 128) {
            dr[m] = (m < KLEN) ? (up[m + 1] - up[m]) : 0.0f;
        }
    }
    __syncthreads();

    const int w    = tid >> 5;
    const int lane = tid & 31;
    const int lr   = lane & 15;   // A row / B,D column
    const int hi   = lane >> 4;   // lane-half selector
    // coef = h^(-alpha) / Gamma(2 - alpha), h = 1/(N-1)
    const float coef = sqrtf(1023.0f) / 0.8862269254527580f;

    float psum = 0.0f;
    // n == 0 residual: frac_lap = 0, u_t[0] = u[1]-u[0]
    if (w == 0 && hi == 0) {
        const float* up = u + (size_t)(rb + lr) * ROWLEN;
        float r0 = up[1] - up[0];
        psum += r0 * r0;
    }

    const float* arow = du_s + lr * PITCH;
    const int    klb  = 2 * hi;              // k_local base for this lane half
    const int    xln  = 1023 - lr + klb;     // lane part of reversed-B index

    // B pair for Toeplitz offset index x (ascending consecutive in LDS)
    auto bpair = [&](int x) {
        v2f bv;
        bv.x = brv[x];
        bv.y = brv[x + 1];
        return bv;
    };
    // residual accumulation for one finished D tile
    auto emit_residual = [&](int n0, const v8f& a8) {
#pragma unroll
        for (int v = 0; v < 8; ++v) {
            const int n = n0 + lr + 1;               // residual column
            if (n < ROWLEN) {
                const int row = rb + v + 8 * hi;     // D-layout row
                const float* up = u + (size_t)row * ROWLEN;
                float ut = (n == ROWLEN - 1)
                               ? (up[ROWLEN - 1] - up[ROWLEN - 2])
                               : 0.5f * (up[n + 1] - up[n - 1]);
                float res = ut - coef * a8[v];
                psum += res * res;
            }
        }
    };

#define FPL_WMMA(AV, BV, ACC)                                             \
    ACC = __builtin_amdgcn_wmma_f32_16x16x4_f32(                          \
        false, AV, false, BV, (short)0, ACC, false, false)

    // wave w owns quads (j0, j0+4, j0+8, j0+12), j0 = w + 16t
    for (int t = 0; t < 4; ++t) {
        const int j0 = w + 16 * t;
        const int n0 = j0 * 16;
        v8f acc0 = {0.f, 0.f, 0.f, 0.f, 0.f, 0.f, 0.f, 0.f};
        v8f acc1 = acc0, acc2 = acc0, acc3 = acc0;

        // shared-A region: i = 0..j0 (j0's diagonal masked by guard zeros)
        for (int i = 0; i <= j0; ++i) {
            const int m0 = i * 16;
            const int x0 = xln - n0 + m0;   // reversed index, tile j0
#pragma unroll
            for (int kk = 0; kk < 16; kk += 4) {
                v2f av = *(const v2f*)(arow + m0 + klb + kk);  // ds_load_b64
                FPL_WMMA(av, bpair(x0 + kk), acc0);
                FPL_WMMA(av, bpair(x0 + kk - 64), acc1);
                FPL_WMMA(av, bpair(x0 + kk - 128), acc2);
                FPL_WMMA(av, bpair(x0 + kk - 192), acc3);
            }
        }
        // tail regions (fixed trip counts -> fully unrolled, B in registers)
        for (int i = j0 + 1; i <= j0 + 4; ++i) {
            const int m0 = i * 16;
            const int x1 = xln - n0 - 64 + m0;
#pragma unroll
            for (int kk = 0; kk < 16; kk += 4) {
                v2f av = *(const v2f*)(arow + m0 + klb + kk);
                FPL_WMMA(av, bpair(x1 + kk), acc1);
                FPL_WMMA(av, bpair(x1 + kk - 64), acc2);
                FPL_WMMA(av, bpair(x1 + kk - 128), acc3);
            }
        }
        for (int i = j0 + 5; i <= j0 + 8; ++i) {
            const int m0 = i * 16;
            const int x2 = xln - n0 - 128 + m0;
#pragma unroll
            for (int kk = 0; kk < 16; kk += 4) {
                v2f av = *(const v2f*)(arow + m0 + klb + kk);
                FPL_WMMA(av, bpair(x2 + kk), acc2);
                FPL_WMMA(av, bpair(x2 + kk - 64), acc3);
            }
        }
        for (int i = j0 + 9; i <= j0 + 12; ++i) {
            const int m0 = i * 16;
            const int x3 = xln - n0 - 192 + m0;
#pragma unroll
            for (int kk = 0; kk < 16; kk += 4) {
                v2f av = *(const v2f*)(arow + m0 + klb + kk);
                FPL_WMMA(av, bpair(x3 + kk), acc3);
            }
        }

        emit_residual(n0, acc0);
        emit_residual(n0 + 64, acc1);
        emit_residual(n0 + 128, acc2);
        emit_residual(n0 + 192, acc3);
    }
#undef FPL_WMMA

    for (int off = 16; off > 0; off >>= 1) psum += __shfl_down(psum, off);
    if (lane == 0) atomicAdd(&acc[1], (double)psum);
}

__global__ void fpl_final_kernel(const double* __restrict__ acc,
                                 float* __restrict__ out) {
    const double inv = 1.0 / (double)NELTS;
    double dl = acc[0] * inv;
    double pl = acc[1] * inv;
    out[0] = (float)(dl + 0.1 * pl);   // total_loss
    out[1] = (float)dl;                // data_loss
    out[2] = (float)pl;                // physics_loss
}

extern "C" void kernel_launch(void* const* d_in, const int* in_sizes, int n_in,
                              void* d_out, int out_size, void* d_ws, size_t ws_size,
                              hipStream_t stream) {
    const float* u_pred = (const float*)d_in[0];
    const float* u_true = (const float*)d_in[1];
    // d_in[2] (x) and d_in[3] (t) are unused by the reference math.
    float*  out = (float*)d_out;
    double* acc = (double*)d_ws;

    const int n       = in_sizes[0];        // B*T*N = 8388608
    const int nrows   = n / ROWLEN;         // 8192
    const int nblocks = nrows / NROW;       // 512
    const size_t smem = (size_t)(NROW * PITCH + BLEN) * sizeof(float);

    fpl_init_kernel<<<1, 1, 0, stream>>>(acc);
    fpl_data_kernel<<<2048, 256, 0, stream>>>(u_pred, u_true, n, acc);
    fpl_physics_kernel<<<nblocks, 128, smem, stream>>>(u_pred, acc);
    fpl_final_kernel<<<1, 1, 0, stream>>>(acc, out);
}